// GIN_BAMultiShapesDataset_59725815218466
// MI455X (gfx1250) — compile-verified
//
#include <hip/hip_runtime.h>

#define N_NODES   100000
#define N_EDGES   3200000
#define N_GRAPHS  1000
#define IN_DIM    10
#define HID       20
#define HID_PAD   32
#define N_CLASSES 2
#define BN_EPS    1e-5f

typedef float v2f __attribute__((ext_vector_type(2)));
typedef float v8f __attribute__((ext_vector_type(8)));

// ---------------------------------------------------------------- utilities
__global__ void zero_f32(float* __restrict__ p, int n) {
    int i = blockIdx.x * blockDim.x + threadIdx.x;
    if (i < n) p[i] = 0.0f;
}

// layer-1: copy x [N x 10] into zero-padded agg [N x 12]
__global__ void copy_pad12(const float* __restrict__ x, float* __restrict__ agg, int n) {
    int i = blockIdx.x * blockDim.x + threadIdx.x;      // over N_NODES*12
    if (i >= n) return;
    int node = i / 12, d = i % 12;
    agg[i] = (d < IN_DIM) ? x[node * IN_DIM + d] : 0.0f;
}

// ------------------------------------------------- edge scatter: agg += h[src]
// DS = stride of source rows, DA = stride of dest rows, D = live features
template <int DS, int DA, int D>
__global__ void scatter_add(const float* __restrict__ h, const int* __restrict__ src,
                            const int* __restrict__ dst, float* __restrict__ agg, int nE) {
    int e = blockIdx.x * blockDim.x + threadIdx.x;
    if (e >= nE) return;
    const float* hp = h + (size_t)src[e] * DS;
    float*       ap = agg + (size_t)dst[e] * DA;
    float v[D];
    if constexpr (D % 4 == 0) {                       // 16B-aligned rows -> b128 loads
#pragma unroll
        for (int i = 0; i < D / 4; ++i) {
            float4 t = ((const float4*)hp)[i];
            v[4*i] = t.x; v[4*i+1] = t.y; v[4*i+2] = t.z; v[4*i+3] = t.w;
        }
    } else {                                          // 8B-aligned rows -> b64 loads
#pragma unroll
        for (int i = 0; i < D / 2; ++i) {
            float2 t = ((const float2*)hp)[i];
            v[2*i] = t.x; v[2*i+1] = t.y;
        }
    }
#pragma unroll
    for (int i = 0; i < D; ++i)
        unsafeAtomicAdd(ap + i, v[i]);                // -> global_atomic_add_f32
}

// ---------------------------------------- WMMA GEMM + bias + BN-stat reduction
// out_pad[N x 32] = A[N x KS (zero-padded)] @ W[K x HID] + bias.
// One wave per 16-row M-tile; computes BOTH 16-col N-tiles (cols 0-31, zero-padded).
// All loads/stores unconditional: padding guarantees zeros, so no EXEC divergence.
template <int K, int KS>     // K = logical reduction dim, KS = padded A stride (%4==0)
__global__ void gemm_bias_stats(const float* __restrict__ A, const float* __restrict__ W,
                                const float* __restrict__ bias, float* __restrict__ out,
                                float* __restrict__ sums, float* __restrict__ sumsq) {
    __shared__ float Wp[KS][HID_PAD];   // zero-padded weights
    __shared__ float bp[HID_PAD];       // zero-padded bias
    for (int i = threadIdx.x; i < KS * HID_PAD; i += blockDim.x) {
        int kk = i >> 5, nn = i & 31;
        Wp[kk][nn] = (kk < K && nn < HID) ? W[kk * HID + nn] : 0.0f;
    }
    if (threadIdx.x < HID_PAD)
        bp[threadIdx.x] = (threadIdx.x < HID) ? bias[threadIdx.x] : 0.0f;
    __syncthreads();

    const int lane = threadIdx.x & 31;
    const int wave = (blockIdx.x * blockDim.x + threadIdx.x) >> 5;
    if (wave >= N_NODES / 16) return;             // wave-uniform: EXEC all-ones at WMMA
    const int m0   = wave * 16;
    const int half = lane >> 4;                   // 0: lanes 0-15, 1: lanes 16-31
    const int l    = lane & 15;

    const float* Arow = A + (size_t)(m0 + l) * KS;
    v8f c0 = {}, c1 = {};
#pragma unroll
    for (int k = 0; k < KS; k += 4) {
        const int ka = k + half * 2;              // f32 16x4 A layout: halves hold K pairs
        v2f a = *(const v2f*)(Arow + ka);         // global_load_b64, unconditional
        v2f b0, b1;
        b0.x = Wp[ka][l];        b0.y = Wp[ka + 1][l];
        b1.x = Wp[ka][16 + l];   b1.y = Wp[ka + 1][16 + l];
        c0 = __builtin_amdgcn_wmma_f32_16x16x4_f32(false, a, false, b0, (short)0, c0, false, false);
        c1 = __builtin_amdgcn_wmma_f32_16x16x4_f32(false, a, false, b1, (short)0, c1, false, false);
    }

    const int   rowBase = m0 + half * 8;          // D layout: lanes>=16 hold rows M=8..15
    const float bv0 = bp[l], bv1 = bp[16 + l];
    float s0 = 0.f, q0 = 0.f, s1 = 0.f, q1 = 0.f;
#pragma unroll
    for (int r = 0; r < 8; ++r) {
        float v0 = c0[r] + bv0;
        float v1 = c1[r] + bv1;
        out[(rowBase + r) * HID_PAD + l]      = v0;   // coalesced, unconditional
        out[(rowBase + r) * HID_PAD + 16 + l] = v1;   // padded cols store exact 0
        s0 += v0; q0 += v0 * v0;
        s1 += v1; q1 += v1 * v1;
    }
    // padded columns contribute exactly 0 -> unconditional commit is safe
    unsafeAtomicAdd(&sums[l], s0);        unsafeAtomicAdd(&sumsq[l], q0);
    unsafeAtomicAdd(&sums[16 + l], s1);   unsafeAtomicAdd(&sumsq[16 + l], q1);
}

// ------------------------------------------------------- BN scale/shift (tiny)
__global__ void finalize_stats(const float* __restrict__ sums, const float* __restrict__ sumsq,
                               const float* __restrict__ g, const float* __restrict__ beta,
                               float* __restrict__ scale, float* __restrict__ shift) {
    int d = threadIdx.x;
    if (d >= HID) return;
    const float invN = 1.0f / (float)N_NODES;
    float mu  = sums[d] * invN;
    float var = sumsq[d] * invN - mu * mu;
    float sc  = g[d] * rsqrtf(var + BN_EPS);
    scale[d]  = sc;
    shift[d]  = beta[d] - mu * sc;
}

// ----------------------- BN + ReLU, write next h and pre-init next-layer agg
__global__ void bn_relu(const float* __restrict__ hlin, const float* __restrict__ scale,
                        const float* __restrict__ shift, float* __restrict__ hout,
                        float* __restrict__ aggout, int total) {
    int i = blockIdx.x * blockDim.x + threadIdx.x;    // over N_NODES*HID
    if (i >= total) return;
    int node = i / HID, d = i % HID;
    float v = fmaxf(hlin[node * HID_PAD + d] * scale[d] + shift[d], 0.0f);
    hout[i]   = v;
    aggout[i] = v;     // (1+eps)*x_i self-term for next layer's aggregation
}

// --------------- last layer: BN + ReLU + global-add-pool into graph embeddings
__global__ void bn_relu_pool(const float* __restrict__ hlin, const float* __restrict__ scale,
                             const float* __restrict__ shift, const int* __restrict__ batch,
                             float* __restrict__ pool, int total) {
    int i = blockIdx.x * blockDim.x + threadIdx.x;
    if (i >= total) return;
    int node = i / HID, d = i % HID;
    float v = fmaxf(hlin[node * HID_PAD + d] * scale[d] + shift[d], 0.0f);
    unsafeAtomicAdd(&pool[(size_t)batch[node] * HID + d], v);
}

// --------------------------------------------------------------- final FC head
__global__ void fc_head(const float* __restrict__ pool, const float* __restrict__ fcW,
                        const float* __restrict__ fcb, float* __restrict__ out) {
    int i = blockIdx.x * blockDim.x + threadIdx.x;
    if (i >= N_GRAPHS * N_CLASSES) return;
    int gph = i / N_CLASSES, cls = i % N_CLASSES;
    float acc = fcb[cls];
#pragma unroll
    for (int d = 0; d < HID; ++d)
        acc += pool[gph * HID + d] * fcW[d * N_CLASSES + cls];
    out[i] = acc;
}

// =============================================================================
extern "C" void kernel_launch(void* const* d_in, const int* in_sizes, int n_in,
                              void* d_out, int out_size, void* d_ws, size_t ws_size,
                              hipStream_t stream) {
    const float* x     = (const float*)d_in[0];
    const int*   ei    = (const int*)d_in[1];
    const int*   src   = ei;
    const int*   dst   = ei + N_EDGES;
    const int*   batch = (const int*)d_in[2];
    const float* W1 = (const float*)d_in[3],  *b1  = (const float*)d_in[4];
    const float* g1 = (const float*)d_in[5],  *be1 = (const float*)d_in[6];
    const float* W2 = (const float*)d_in[7],  *b2  = (const float*)d_in[8];
    const float* g2 = (const float*)d_in[9],  *be2 = (const float*)d_in[10];
    const float* W3 = (const float*)d_in[11], *b3  = (const float*)d_in[12];
    const float* g3 = (const float*)d_in[13], *be3 = (const float*)d_in[14];
    const float* fcW = (const float*)d_in[15], *fcb = (const float*)d_in[16];
    float* out = (float*)d_out;

    // ---- workspace carve-up (bytes) ----
    char* ws = (char*)d_ws;
    float* agg   = (float*)(ws);                          // N*20 (layer1 uses N*12)
    float* hlin  = (float*)(ws + 8000000);                // N*32 padded
    float* h     = (float*)(ws + 20800000);               // N*20
    float* pool  = (float*)(ws + 28800000);               // N_GRAPHS*20
    float* sums  = (float*)(ws + 28880000);               // 32
    float* sumsq = sums + 32;
    float* scale = sums + 64;
    float* shift = sums + 96;

    const int TB = 256;
    const int edgeBlocks = (N_EDGES + TB - 1) / TB;
    const int nhTotal    = N_NODES * HID;
    const int nhBlocks   = (nhTotal + TB - 1) / TB;
    const int gemmWaves  = N_NODES / 16;                  // one wave per 16-row tile
    const int gemmBlocks = (gemmWaves * 32 + TB - 1) / TB;

    // ================= layer 1 (K = 10, A stride 12 zero-padded) =================
    {
        const int n1 = N_NODES * 12;
        copy_pad12<<<(n1 + TB - 1) / TB, TB, 0, stream>>>(x, agg, n1);
        zero_f32<<<1, 64, 0, stream>>>(sums, 64);
        scatter_add<IN_DIM, 12, IN_DIM><<<edgeBlocks, TB, 0, stream>>>(x, src, dst, agg, N_EDGES);
        gemm_bias_stats<IN_DIM, 12><<<gemmBlocks, TB, 0, stream>>>(agg, W1, b1, hlin, sums, sumsq);
        finalize_stats<<<1, 32, 0, stream>>>(sums, sumsq, g1, be1, scale, shift);
        bn_relu<<<nhBlocks, TB, 0, stream>>>(hlin, scale, shift, h, agg, nhTotal);
    }
    // ================= layer 2 (K = 20) =================
    {
        zero_f32<<<1, 64, 0, stream>>>(sums, 64);
        scatter_add<HID, HID, HID><<<edgeBlocks, TB, 0, stream>>>(h, src, dst, agg, N_EDGES);
        gemm_bias_stats<HID, HID><<<gemmBlocks, TB, 0, stream>>>(agg, W2, b2, hlin, sums, sumsq);
        finalize_stats<<<1, 32, 0, stream>>>(sums, sumsq, g2, be2, scale, shift);
        bn_relu<<<nhBlocks, TB, 0, stream>>>(hlin, scale, shift, h, agg, nhTotal);
    }
    // ================= layer 3 + pooling =================
    {
        zero_f32<<<1, 64, 0, stream>>>(sums, 64);
        zero_f32<<<(N_GRAPHS * HID + TB - 1) / TB, TB, 0, stream>>>(pool, N_GRAPHS * HID);
        scatter_add<HID, HID, HID><<<edgeBlocks, TB, 0, stream>>>(h, src, dst, agg, N_EDGES);
        gemm_bias_stats<HID, HID><<<gemmBlocks, TB, 0, stream>>>(agg, W3, b3, hlin, sums, sumsq);
        finalize_stats<<<1, 32, 0, stream>>>(sums, sumsq, g3, be3, scale, shift);
        bn_relu_pool<<<nhBlocks, TB, 0, stream>>>(hlin, scale, shift, batch, pool, nhTotal);
    }
    // ================= FC head =================
    fc_head<<<(N_GRAPHS * N_CLASSES + TB - 1) / TB, TB, 0, stream>>>(pool, fcW, fcb, out);
}